// LabelSmoothing_41635412967981
// MI455X (gfx1250) — compile-verified
//
#include <hip/hip_runtime.h>

#define VOCAB   32000
#define NTOK    4096
#define ROW_F4  (VOCAB / 4)      // 8000 float4 per row
#define PADIDX  0

typedef __attribute__((ext_vector_type(4))) float v4f;
typedef __attribute__((ext_vector_type(2))) float v2f;
typedef __attribute__((ext_vector_type(8))) float v8f;

// Stage 1: one wave per row. Streams the row with non-temporal b128 loads,
// accumulates per-lane partials, then reduces across the wave with a single
// V_WMMA_F32_16X16X4_F32 (B = ones => D[m][n] = sum_k A[m][k]; tile sum is
// layout-agnostic). Lane 0 finishes the closed-form label-smoothing loss.
__global__ __launch_bounds__(256) void ls_rowloss_kernel(
    const float* __restrict__ X,       // [NTOK, VOCAB] log-probs
    const int*   __restrict__ tgt,     // [NTOK]
    float*       __restrict__ rowloss) // [NTOK] per-row loss
{
    const int lane = threadIdx.x & 31;
    const int wave = threadIdx.x >> 5;
    const int row  = blockIdx.x * 8 + wave;

    const v4f* __restrict__ p = (const v4f*)(X + (size_t)row * VOCAB);

    // Two independent accumulators -> two outstanding b128 loads per iter.
    v4f acc0 = {0.f, 0.f, 0.f, 0.f};
    v4f acc1 = {0.f, 0.f, 0.f, 0.f};
    // 8000 float4 / (64 per wave-iter) = 125 uniform iterations, full EXEC.
    for (int i = lane; i < ROW_F4; i += 64) {
        v4f u = __builtin_nontemporal_load(&p[i]);        // TH=NT: single-pass stream
        v4f w = __builtin_nontemporal_load(&p[i + 32]);
        acc0 += u;
        acc1 += w;
    }

    // Pack per-lane partials into the 2 A-matrix VGPRs (layout irrelevant for
    // a tile sum) and reduce across the wave with one f32 WMMA.
    v2f a;
    a[0] = (acc0[0] + acc0[1]) + (acc1[0] + acc1[1]);
    a[1] = (acc0[2] + acc0[3]) + (acc1[2] + acc1[3]);
    v2f b = {1.0f, 1.0f};                 // B tile = all ones
    v8f c = {0.f, 0.f, 0.f, 0.f, 0.f, 0.f, 0.f, 0.f};
    c = __builtin_amdgcn_wmma_f32_16x16x4_f32(
            /*neg_a=*/false, a, /*neg_b=*/false, b,
            /*c_mod=*/(short)0, c, /*reuse_a=*/false, /*reuse_b=*/false);

    // Lanes 0-15 hold D[m=0..7][n], lanes 16-31 hold D[m=8..15][n]; columns
    // are identical, so full tile sum = local vgpr-sum + xor-16 partner.
    float s = ((c[0] + c[1]) + (c[2] + c[3])) + ((c[4] + c[5]) + (c[6] + c[7]));
    float S = s + __shfl_xor(s, 16, 32);   // row sum, valid on all lanes

    if (lane == 0) {
        const float BASE = 0.1f / (float)(VOCAB - 1);
        const float CONF = 1.0f - 0.1f;
        const int   t    = tgt[row];
        const float x0   = X[(size_t)row * VOCAB + PADIDX];
        const float xt   = X[(size_t)row * VOCAB + t];
        const float K0   = BASE * (float)(VOCAB - 2) * logf(BASE)
                         + CONF * logf(CONF);
        const float loss = K0 - BASE * S + BASE * x0 + (BASE - CONF) * xt;
        rowloss[row] = (t != PADIDX) ? loss : 0.0f;
    }
}

// Stage 2: deterministic single-block tree reduction of 4096 row losses.
__global__ __launch_bounds__(256) void ls_reduce_kernel(
    const float* __restrict__ rowloss, float* __restrict__ out)
{
    __shared__ float sm[256];
    float s = 0.0f;
    for (int i = threadIdx.x; i < NTOK; i += 256)   // fixed order: 16 adds/thread
        s += rowloss[i];
    sm[threadIdx.x] = s;
    __syncthreads();
    for (int off = 128; off > 0; off >>= 1) {
        if ((int)threadIdx.x < off)
            sm[threadIdx.x] += sm[threadIdx.x + off];
        __syncthreads();
    }
    if (threadIdx.x == 0)
        out[0] = sm[0];
}

extern "C" void kernel_launch(void* const* d_in, const int* in_sizes, int n_in,
                              void* d_out, int out_size, void* d_ws, size_t ws_size,
                              hipStream_t stream) {
    (void)in_sizes; (void)n_in; (void)out_size; (void)ws_size;
    const float* X   = (const float*)d_in[0];   // model_output [4096, 32000] f32
    const int*   tgt = (const int*)  d_in[1];   // target [4096]
    float*       out = (float*)d_out;           // scalar f32
    float*       rowloss = (float*)d_ws;        // 4096 f32 = 16 KB scratch

    ls_rowloss_kernel<<<dim3(NTOK / 8), dim3(256), 0, stream>>>(X, tgt, rowloss);
    ls_reduce_kernel<<<dim3(1), dim3(256), 0, stream>>>(rowloss, out);
}